// Pitch_33071248179731
// MI455X (gfx1250) — compile-verified
//
#include <hip/hip_runtime.h>
#include <cmath>

typedef __attribute__((ext_vector_type(2))) float v2f;
typedef __attribute__((ext_vector_type(8))) float v8f;

#define WSZ   2048      // frame length W
#define TAU   2048      // tau_max
#define STEP  256       // w_step
#define NBAT  16        // B
#define NMIDI 80        // output lags
#define PADL  16        // left zero pad (covers a[k-n], n<=15)
#define BUFLEN (PADL + WSZ + 256)   // 2320; right zeros cover max read idx 2303

// Block-wide (256-thread) inclusive scan of one value per thread.
// Wave32 shfl scan + 8-entry wave-sum combine. Two barriers total.
__device__ __forceinline__ float block_scan_incl(float v, int tid, float* wsum8) {
    const int lane = tid & 31;
    const int wave = tid >> 5;
    float s = v;
    #pragma unroll
    for (int d = 1; d < 32; d <<= 1) {
        float t = __shfl_up(s, d, 32);
        if (lane >= d) s += t;
    }
    __syncthreads();                  // protect wsum8 against previous readers
    if (lane == 31) wsum8[wave] = s;  // wave totals
    __syncthreads();
    float off = 0.0f;
    #pragma unroll
    for (int w = 0; w < 7; ++w)
        off += (w < wave) ? wsum8[w] : 0.0f;
    return s + off;
}

__launch_bounds__(256)
__global__ void yin_wmma_kernel(const float* __restrict__ x,
                                const int*  __restrict__ sr_p,
                                float* __restrict__ out,
                                int t, int F) {
    __shared__ float buf[BUFLEN];     // [0..15]=0, [16+s]=f[s] (s<W), tail 0
    __shared__ float P[16 * 256];     // per-unit partial D tiles
    __shared__ float r[TAU];          // autocorr, then df in place
    __shared__ float cs[WSZ + 1];     // prefix of f^2
    __shared__ float S[TAU];          // inclusive prefix of df
    __shared__ float wsum[8];

    const int tid   = threadIdx.x;
    const int frame = blockIdx.x;
    const int b     = blockIdx.y;
    const long base = (long)b * (long)t + (long)frame * STEP;

    // ---- Phase 1: stage padded frame into LDS (float4) -----------------------
    {
        const float4 z4 = make_float4(0.f, 0.f, 0.f, 0.f);
        const float4* xv = (const float4*)(x + base);   // base is 256-float aligned
        for (int i = tid; i < WSZ / 4; i += 256)
            *(float4*)&buf[PADL + 4 * i] = xv[i];
        for (int i = tid; i < 68; i += 256) {           // 4 left + 64 right pad vecs
            int idx = (i < 4) ? (4 * i) : (PADL + WSZ + 4 * (i - 4));
            *(float4*)&buf[idx] = z4;
        }
    }
    __syncthreads();

    // ---- Phase 2: autocorrelation via V_WMMA_F32_16X16X4_F32 -----------------
    // tau = 256*mt + 16*m + n ; D[m][n] = sum_k a[k-n] * b[k + 256*mt + 16*m]
    // 16 balanced units: unit u = (tile mt=u>>1, K-half h=u&1); wave w gets {w, 15-w}.
    {
        const int lane  = tid & 31;
        const int wave  = __builtin_amdgcn_readfirstlane(tid >> 5); // SGPR => scalar loops
        const int lm    = lane & 15;         // M (A rows) / N (B,D cols)
        const int hiK   = (lane >> 4) << 1;  // lanes 0-15: K={0,1}; 16-31: K={2,3}
        const int mbase = (lane < 16) ? 0 : 8;

        #pragma unroll
        for (int uu = 0; uu < 2; ++uu) {
            const int u    = uu ? (15 - wave) : wave;
            const int mt   = u >> 1;
            const int h    = u & 1;
            const int Kt   = WSZ - 256 * mt;          // triangular K extent
            const int kbeg = h * (Kt >> 1);
            const int kend = kbeg + (Kt >> 1);
            const int aoff = PADL + hiK + 256 * mt + 16 * lm;  // A[m][k]=b[k+256mt+16m]
            const int boff = PADL + hiK - lm;                  // B[k][n]=a[k-n]

            v8f acc = {0.f,0.f,0.f,0.f,0.f,0.f,0.f,0.f};
            #pragma unroll 4
            for (int k = kbeg; k < kend; k += 4) {
                v2f av, bv;
                av.x = buf[aoff + k];
                av.y = buf[aoff + k + 1];
                bv.x = buf[boff + k];
                bv.y = buf[boff + k + 1];
                acc = __builtin_amdgcn_wmma_f32_16x16x4_f32(
                          false, av, false, bv, (short)0, acc, false, false);
            }
            // D layout: VGPR q -> M=q (lanes 0-15) / q+8 (lanes 16-31), N=lane%16
            #pragma unroll
            for (int q = 0; q < 8; ++q)
                P[u * 256 + 16 * (q + mbase) + lm] = acc[q];
        }
    }
    __syncthreads();

    // combine the two K-halves of each tile: r[256*mt + e] = P[2mt][e] + P[2mt+1][e]
    for (int i = tid; i < TAU; i += 256) {
        int mt = i >> 8, e = i & 255;
        r[i] = P[(2 * mt) * 256 + e] + P[(2 * mt + 1) * 256 + e];
    }
    __syncthreads();

    // ---- Phase 3a: cs = prefix sum of f^2 ------------------------------------
    {
        float loc[8];
        float s = 0.0f;
        const int b8 = tid * 8;
        #pragma unroll
        for (int j = 0; j < 8; ++j) {
            float v = buf[PADL + b8 + j];
            loc[j] = v * v;
            s += loc[j];
        }
        float run = block_scan_incl(s, tid, wsum) - s;   // exclusive prefix
        if (tid == 0) cs[0] = 0.0f;
        #pragma unroll
        for (int j = 0; j < 8; ++j) { run += loc[j]; cs[b8 + j + 1] = run; }
    }
    __syncthreads();

    // ---- Phase 3b: df[tau] in place over r -----------------------------------
    {
        const float E = cs[WSZ];
        for (int tau = tid; tau < TAU; tau += 256) {
            float df = cs[WSZ - tau] + E - cs[tau] - 2.0f * r[tau];
            r[tau] = df;   // same-thread read+write: safe
        }
    }
    __syncthreads();

    // ---- Phase 3c: S = inclusive prefix sum of df ----------------------------
    {
        float loc[8];
        float s = 0.0f;
        const int b8 = tid * 8;
        #pragma unroll
        for (int j = 0; j < 8; ++j) { loc[j] = r[b8 + j]; s += loc[j]; }
        float run = block_scan_incl(s, tid, wsum) - s;
        #pragma unroll
        for (int j = 0; j < 8; ++j) { run += loc[j]; S[b8 + j] = run; }
    }
    __syncthreads();

    // ---- Phase 4: interpolate cmndf at 80 midi lags --------------------------
    if (tid < NMIDI) {
        const int sr = *sr_p;
        const float S0 = S[0];  // exclude df[0], matching cumsum(df[1:])
        auto cmndf_at = [&](int idx) -> float {
            if (idx <= 0) return 1.0f;
            return r[idx] * (float)idx / ((S[idx] - S0) + 1e-8f);
        };
        const int midi = 5 + tid;
        // float64 lag computation so floor/ceil indices match numpy
        double cm = (double)sr / (440.0 * exp2(((double)(midi - 69)) / 12.0));
        int   fl   = (int)floor(cm);
        int   ce   = (int)ceil(cm);
        float frac = (float)(cm - (double)fl);
        float den  = (float)(ce - fl);
        float lo = cmndf_at(fl);
        float hi = cmndf_at(ce);
        float y  = (den != 0.0f) ? ((hi - lo) / den * frac + lo) : lo;
        out[((long)b * NMIDI + tid) * (long)F + frame] = y;
    }
}

extern "C" void kernel_launch(void* const* d_in, const int* in_sizes, int n_in,
                              void* d_out, int out_size, void* d_ws, size_t ws_size,
                              hipStream_t stream) {
    const float* x    = (const float*)d_in[0];
    const int*   sr_p = (const int*)d_in[3];   // sr scalar (device)
    float*       out  = (float*)d_out;

    const int t = in_sizes[0] / NBAT;                    // 262144
    const int F = (t - WSZ + STEP - 1) / STEP;           // 1016 frames

    dim3 grid(F, NBAT);
    dim3 block(256);
    yin_wmma_kernel<<<grid, block, 0, stream>>>(x, sr_p, out, t, F);
}